// DeformConv2d_10307921511064
// MI455X (gfx1250) — compile-verified
//
#include <hip/hip_runtime.h>
#include <math.h>
#include <stdint.h>

// Problem constants (from reference)
#define Bn   8
#define Hh   128
#define Ww   128
#define HW   (Hh*Ww)
#define Cg   32      // channels per group
#define IND  64
#define OUTD 64

typedef float v2f __attribute__((ext_vector_type(2)));
typedef float v8f __attribute__((ext_vector_type(8)));
typedef unsigned int u32x4 __attribute__((ext_vector_type(4)));
typedef int i32x8 __attribute__((ext_vector_type(8)));
typedef int i32x4 __attribute__((ext_vector_type(4)));

// ---------------------------------------------------------------------------
// Tensor Data Mover: DMA a [rows x kfloats] fp32 matrix from global into LDS
// (lds byte offset 0 = start of dynamic LDS). Descriptor per CDNA5 ISA §8:
//  group0: count=1 | lds_addr | global_addr[56:0] | type=2
//  group1: data_size=2 (4B), tensor_dim0=kfloats, tensor_dim1=rows,
//          tile_dim0=kfloats, tile_dim1=rows, tensor_dim0_stride=kfloats
// Issued by one wave; completion via s_wait_tensorcnt 0 (TDM ignores EXEC).
// This toolchain's builtin takes 6 args (g0, g1, g2, g3, extra, cpol).
// ---------------------------------------------------------------------------
__device__ __forceinline__ void tdm_load_weights(const float* gptr,
                                                 unsigned kfloats, unsigned rows)
{
    unsigned long long ga = (unsigned long long)(uintptr_t)gptr;
    u32x4 g0;
    g0.x = 1u;                                                  // count=1 (valid user D#)
    g0.y = 0u;                                                  // lds_addr (bytes)
    g0.z = (unsigned)(ga & 0xFFFFFFFFull);                      // global_addr[31:0]
    g0.w = (unsigned)((ga >> 32) & 0x1FFFFFFull) | (2u << 30);  // addr[56:32] | type=2

    i32x8 g1;
    g1[0] = (int)(2u << 16);                                    // data_size=2 (4 bytes)
    g1[1] = (int)((kfloats & 0xFFFFu) << 16);                   // tensor_dim0[15:0] @63:48
    g1[2] = (int)((kfloats >> 16) | ((rows & 0xFFFFu) << 16));  // dim0[31:16] | dim1[15:0]
    g1[3] = (int)((rows >> 16) | ((kfloats & 0xFFFFu) << 16));  // dim1[31:16] | tile_dim0
    g1[4] = (int)(rows & 0xFFFFu);                              // tile_dim1 | tile_dim2=0
    g1[5] = (int)kfloats;                                       // tensor_dim0_stride[31:0]
    g1[6] = 0;                                                  // stride[47:32] | dim1_stride lo
    g1[7] = 0;

    i32x4 z4 = {0, 0, 0, 0};                                    // 2D: groups 2/3 unused
    i32x8 z8 = {0, 0, 0, 0, 0, 0, 0, 0};
    __builtin_amdgcn_tensor_load_to_lds(g0, g1, z4, z4, z8, 0);
}

// ---------------------------------------------------------------------------
// Offset-predictor 3x3 conv (32 -> OC channels), direct fp32 FMA.
// p layout: [B][OC][H][W] raw conv output (sigmoid applied later in deform).
// ---------------------------------------------------------------------------
template<int OC>
__global__ void offset_conv_kernel(const float* __restrict__ x,
                                   const float* __restrict__ w,
                                   const float* __restrict__ bias,
                                   float* __restrict__ p, int goff)
{
    int idx = blockIdx.x * blockDim.x + threadIdx.x;
    const int total = Bn * OC * HW;
    if (idx >= total) return;
    int xw = idx & (Ww - 1);
    int t  = idx >> 7;            // / Ww
    int yh = t & (Hh - 1);
    t >>= 7;                      // / Hh
    int oc = t % OC;
    int b  = t / OC;

    const float* xp = x + (size_t)b * IND * HW + (size_t)goff * HW;
    const float* wp = w + (size_t)oc * Cg * 9;
    float acc = bias[oc];
    #pragma unroll
    for (int kh = 0; kh < 3; ++kh) {
        int sy = yh + kh - 1;
        if ((unsigned)sy >= (unsigned)Hh) continue;
        #pragma unroll
        for (int kw = 0; kw < 3; ++kw) {
            int sx = xw + kw - 1;
            if ((unsigned)sx >= (unsigned)Ww) continue;
            const float* xq = xp + sy * Ww + sx;
            const float* wq = wp + kh * 3 + kw;
            #pragma unroll
            for (int c = 0; c < Cg; ++c)
                acc = fmaf(xq[(size_t)c * HW], wq[(size_t)c * 9], acc);
        }
    }
    p[idx] = acc;
}

// ---------------------------------------------------------------------------
// Fused modulated deformable conv, one wave = 16 pixels (N-tile), M=32.
// Weights TDM-staged in LDS; A-fragments from ds_load, B from bilinear
// gathers (L2-resident x). V_WMMA_F32_16X16X4_F32, two M-tiles per wave.
// A 16x4 layout: lane m=lane&15, {a.x,a.y} = A[m][kb+2*(lane>=16) + {0,1}]
// B 4x16 layout: lane n=lane&15, {b.x,b.y} = B[kb+2*(lane>=16) + {0,1}][n]
// C/D 16x16:     lane, vgpr v -> row v+8*(lane>=16), col lane&15
// ---------------------------------------------------------------------------
template<int KD, int PAD>
__global__ void deform_wmma_kernel(const float* __restrict__ x,
                                   const float* __restrict__ p,
                                   const float* __restrict__ wt,   // [32][32][KD][KD]
                                   const float* __restrict__ bias, // [32]
                                   float* __restrict__ y, int goff)
{
    constexpr int KK  = KD * KD;
    constexpr int PCH = 3 * KK;
    constexpr int CKK = Cg * KK;

    extern __shared__ float wlds[];                 // [32][Cg*KK] staged weights

    // --- TDM weight staging (wave 0 issues; all waves sync) ---
    if ((threadIdx.x >> 5) == 0) {
        tdm_load_weights(wt, (unsigned)CKK, (unsigned)Cg);
        __builtin_amdgcn_s_wait_tensorcnt(0);
    }
    __syncthreads();

    const int lane  = threadIdx.x & 31;
    const int wv    = blockIdx.x * (blockDim.x >> 5) + (threadIdx.x >> 5);
    const int xb    = (wv & 7) * 16;       // 8 x-tiles per row
    int t = wv >> 3;
    const int yy    = t & (Hh - 1);
    const int b     = t >> 7;
    const int n     = lane & 15;           // pixel within tile / A row / D col
    const int khalf = lane >> 4;           // K sub-phase per WMMA layout
    const int px_x  = xb + n;

    const float* xp = x + (size_t)b * IND * HW + (size_t)goff * HW;
    const float* pp = p + (size_t)b * PCH * HW + (size_t)yy * Ww + px_x;
    float*       yp = y + (size_t)b * OUTD * HW + (size_t)goff * HW
                        + (size_t)yy * Ww + xb;

    v8f acc0 = {}; v8f acc1 = {};

    for (int kk = 0; kk < KK; ++kk) {
        const int iy = kk / KD, jx = kk % KD;
        float oh = pp[(size_t)kk * HW];
        float ow = pp[(size_t)(KK + kk) * HW];
        float mk = 1.0f / (1.0f + __expf(-pp[(size_t)(2 * KK + kk) * HW]));
        float py = (float)(yy - PAD + iy) + oh;
        float px = (float)(px_x - PAD + jx) + ow;
        float y0f = floorf(py), x0f = floorf(px);
        float wy = py - y0f, wx = px - x0f;
        int y0 = (int)y0f, x0 = (int)x0f;
        bool vy0 = (y0 >= 0) && (y0 < Hh);
        bool vy1 = (y0 + 1 >= 0) && (y0 + 1 < Hh);
        bool vx0 = (x0 >= 0) && (x0 < Ww);
        bool vx1 = (x0 + 1 >= 0) && (x0 + 1 < Ww);
        int cy0 = min(max(y0, 0), Hh - 1) * Ww;
        int cy1 = min(max(y0 + 1, 0), Hh - 1) * Ww;
        int cx0 = min(max(x0, 0), Ww - 1);
        int cx1 = min(max(x0 + 1, 0), Ww - 1);
        float w00 = (1.0f - wy) * (1.0f - wx) * mk * ((vy0 && vx0) ? 1.0f : 0.0f);
        float w01 = (1.0f - wy) * wx          * mk * ((vy0 && vx1) ? 1.0f : 0.0f);
        float w10 = wy * (1.0f - wx)          * mk * ((vy1 && vx0) ? 1.0f : 0.0f);
        float w11 = wy * wx                   * mk * ((vy1 && vx1) ? 1.0f : 0.0f);
        int i00 = cy0 + cx0, i01 = cy0 + cx1, i10 = cy1 + cx0, i11 = cy1 + cx1;

        #pragma unroll
        for (int kc = 0; kc < Cg; kc += 4) {
            int c0 = kc + khalf * 2, c1 = c0 + 1;
            const float* x0p = xp + (size_t)c0 * HW;
            const float* x1p = xp + (size_t)c1 * HW;
            v2f bfrag;
            bfrag.x = w00 * x0p[i00] + w01 * x0p[i01] + w10 * x0p[i10] + w11 * x0p[i11];
            bfrag.y = w00 * x1p[i00] + w01 * x1p[i01] + w10 * x1p[i10] + w11 * x1p[i11];

            const float* wr0 = wlds + (size_t)n * CKK + kk;       // row n (LDS)
            const float* wr1 = wr0 + (size_t)16 * CKK;            // row n+16 (LDS)
            v2f a0, a1;
            a0.x = wr0[(size_t)c0 * KK]; a0.y = wr0[(size_t)c1 * KK];
            a1.x = wr1[(size_t)c0 * KK]; a1.y = wr1[(size_t)c1 * KK];

            acc0 = __builtin_amdgcn_wmma_f32_16x16x4_f32(false, a0, false, bfrag,
                                                         (short)0, acc0, false, false);
            acc1 = __builtin_amdgcn_wmma_f32_16x16x4_f32(false, a1, false, bfrag,
                                                         (short)0, acc1, false, false);
        }
    }

    #pragma unroll
    for (int v = 0; v < 8; ++v) {
        int r0 = v + 8 * khalf;
        yp[(size_t)r0 * HW + n]        = acc0[v] + bias[r0];
        yp[(size_t)(16 + r0) * HW + n] = acc1[v] + bias[16 + r0];
    }
}

// ---------------------------------------------------------------------------
// Final 3x3 conv 64 -> 64 as im2col GEMM: M=64 (4 WMMA M-tiles), K=576 swept
// tap-major in steps of 4, N-tile of 16 pixels per wave. Weights TDM-staged.
// ---------------------------------------------------------------------------
__global__ void final_conv_wmma_kernel(const float* __restrict__ y,
                                       const float* __restrict__ w,    // [64][64][3][3]
                                       const float* __restrict__ bias, // [64]
                                       float* __restrict__ out)
{
    constexpr int KTOT = IND * 9;                   // 576
    extern __shared__ float wlds[];                 // [64][576]

    if ((threadIdx.x >> 5) == 0) {
        tdm_load_weights(w, (unsigned)KTOT, (unsigned)OUTD);
        __builtin_amdgcn_s_wait_tensorcnt(0);
    }
    __syncthreads();

    const int lane  = threadIdx.x & 31;
    const int wv    = blockIdx.x * (blockDim.x >> 5) + (threadIdx.x >> 5);
    const int xb    = (wv & 7) * 16;
    int t = wv >> 3;
    const int yy    = t & (Hh - 1);
    const int b     = t >> 7;
    const int n     = lane & 15;
    const int khalf = lane >> 4;

    const float* yp = y + (size_t)b * IND * HW;
    v8f acc[4] = {};

    #pragma unroll
    for (int kk = 0; kk < 9; ++kk) {
        int dy = kk / 3 - 1, dx = kk % 3 - 1;
        int sy = yy + dy;
        int sx = xb + n + dx;
        bool valid = ((unsigned)sy < (unsigned)Hh) && ((unsigned)sx < (unsigned)Ww);
        int idx = min(max(sy, 0), Hh - 1) * Ww + min(max(sx, 0), Ww - 1);
        float vm = valid ? 1.0f : 0.0f;

        for (int kc = 0; kc < IND; kc += 4) {
            int c0 = kc + khalf * 2, c1 = c0 + 1;
            v2f bfrag;
            bfrag.x = yp[(size_t)c0 * HW + idx] * vm;
            bfrag.y = yp[(size_t)c1 * HW + idx] * vm;
            #pragma unroll
            for (int mt = 0; mt < 4; ++mt) {
                const float* wr = wlds + (size_t)(mt * 16 + n) * KTOT + kk;  // LDS
                v2f a;
                a.x = wr[(size_t)c0 * 9];
                a.y = wr[(size_t)c1 * 9];
                acc[mt] = __builtin_amdgcn_wmma_f32_16x16x4_f32(false, a, false, bfrag,
                                                                (short)0, acc[mt], false, false);
            }
        }
    }

    float* op = out + (size_t)b * OUTD * HW + (size_t)yy * Ww + xb;
    #pragma unroll
    for (int mt = 0; mt < 4; ++mt) {
        #pragma unroll
        for (int v = 0; v < 8; ++v) {
            int r = mt * 16 + v + 8 * khalf;
            op[(size_t)r * HW + n] = acc[mt][v] + bias[r];
        }
    }
}

// ---------------------------------------------------------------------------
extern "C" void kernel_launch(void* const* d_in, const int* in_sizes, int n_in,
                              void* d_out, int out_size, void* d_ws, size_t ws_size,
                              hipStream_t stream)
{
    const float* x       = (const float*)d_in[0];
    const float* pg1_w   = (const float*)d_in[1];
    const float* pg1_b   = (const float*)d_in[2];
    const float* pg2_w   = (const float*)d_in[3];
    const float* pg2_b   = (const float*)d_in[4];
    const float* weight1 = (const float*)d_in[5];
    const float* weight2 = (const float*)d_in[6];
    const float* dcn_b   = (const float*)d_in[7];
    const float* pw_w    = (const float*)d_in[8];
    const float* pw_b    = (const float*)d_in[9];
    float* out = (float*)d_out;

    // Workspace partition: p1 (B*27*HW), p2 (B*75*HW), y (B*64*HW) — ~87 MB
    float* p1   = (float*)d_ws;
    float* p2   = p1 + (size_t)Bn * 27 * HW;
    float* ybuf = p2 + (size_t)Bn * 75 * HW;

    {
        int total = Bn * 27 * HW;
        offset_conv_kernel<27><<<(total + 255) / 256, 256, 0, stream>>>(x, pg1_w, pg1_b, p1, 0);
    }
    {
        int total = Bn * 75 * HW;
        offset_conv_kernel<75><<<(total + 255) / 256, 256, 0, stream>>>(x, pg2_w, pg2_b, p2, 32);
    }

    const int tiles = Bn * Hh * (Ww / 16);   // 8192 waves, 4 waves/block
    // Dynamic LDS: full weight matrix per block (36 KB / 100 KB / 144 KB,
    // all within the 320 KB WGP LDS budget).
    deform_wmma_kernel<3, 1><<<tiles / 4, 128, (size_t)Cg * Cg * 9  * 4, stream>>>
        (x, p1, weight1, dcn_b, ybuf, 0);
    deform_wmma_kernel<5, 2><<<tiles / 4, 128, (size_t)Cg * Cg * 25 * 4, stream>>>
        (x, p2, weight2, dcn_b, ybuf, 32);
    final_conv_wmma_kernel<<<tiles / 4, 128, (size_t)OUTD * IND * 9 * 4, stream>>>
        (ybuf, pw_w, pw_b, out);
}